// DiagonalMemoryOperator_78288663871710
// MI455X (gfx1250) — compile-verified
//
#include <hip/hip_runtime.h>
#include <stdint.h>

// out[n, d] = x[n, d] * (-|W[d, d]|)
// Pure HBM streaming: 1 GiB moved, 0.27 GFLOP -> 23.3 TB/s roofline, ~46 us
// floor. No WMMA by design (would add 16x wasted MACs + tile-shaped traffic
// to an elementwise op). Optimization = minimal-issue NT b128 stream.

typedef float v4f __attribute__((ext_vector_type(4)));

#define DIM      2048
#define ROW_V4   (DIM / 4)          // 512 float4 per row (power of two)
#define ROW_MASK (ROW_V4 - 1)       // 511
#define TPB      256
#define BLOCKS   2048               // stride = BLOCKS*TPB = 524288 v4,
                                    // an exact multiple of ROW_V4 (see below)

// ---------------------------------------------------------------------------
// Kernel 1: lam[c] = -fabs(W[c, c])   (2048 stride-2049 loads, done once)
// ---------------------------------------------------------------------------
__global__ void lam_gather_kernel(const float* __restrict__ W,
                                  float* __restrict__ lam) {
    int c = blockIdx.x * blockDim.x + threadIdx.x;
    if (c < DIM) {
        lam[c] = -fabsf(W[(size_t)c * (DIM + 1)]);
    }
}

// ---------------------------------------------------------------------------
// Kernel 2: out = x * lam[col], streamed.
//   - lam staged global->LDS with GLOBAL_LOAD_ASYNC_TO_LDS_B128 (CDNA5 async
//     DMA path, ASYNCcnt): no VGPR round-trip for the staging copy.
//     NOTE: the instruction offset applies to BOTH global and LDS addresses
//     so one base + offset:4096 copies
//     the second 4 KB half.
//   - Grid stride (524288 v4) is a multiple of the row width (512 v4), so
//     each thread's column -- and therefore its lam float4 -- is LOOP
//     INVARIANT: read once from LDS, held in registers.
//   - Hot loop is then just: NT global_load_b128 -> 4x v_mul -> NT
//     global_store_b128 -> 32-bit index add. Minimal issue per 16 bytes.
// ---------------------------------------------------------------------------
__global__ __launch_bounds__(TPB) void diag_scale_kernel(
        const float* __restrict__ x,
        const float* __restrict__ lam,
        float* __restrict__ out,
        unsigned n_v4) {
    __shared__ float s_lam[DIM];   // 8 KB

    {
        unsigned t = threadIdx.x;
        // Flat->LDS: hardware LDS address is addr[31:0] of the flat pointer.
        unsigned lds_addr = (unsigned)(uintptr_t)(void*)&s_lam[0] + t * 16u;
        const float* g = lam + (size_t)t * 4u;
        asm volatile(
            "global_load_async_to_lds_b128 %0, %1, off\n\t"
            "global_load_async_to_lds_b128 %0, %1, off offset:4096"
            :: "v"(lds_addr), "v"(g)
            : "memory");
        // Our async LDS writes must land before we signal the barrier.
        asm volatile("s_wait_asynccnt 0x0" ::: "memory");
    }
    __syncthreads();

    const v4f* __restrict__ xv = (const v4f*)x;
    v4f* __restrict__ ov = (v4f*)out;
    const v4f* __restrict__ sl = (const v4f*)s_lam;

    const unsigned tid    = blockIdx.x * TPB + threadIdx.x;
    const unsigned stride = gridDim.x * TPB;     // multiple of ROW_V4 by launch

    // Column index is invariant across the grid-stride loop.
    const v4f l = sl[tid & ROW_MASK];

    unsigned i = tid;
    // 4-way unrolled main loop: 4 independent NT b128 loads in flight.
    for (; i + 3u * stride < n_v4; i += 4u * stride) {
        v4f a0 = __builtin_nontemporal_load(&xv[i]);
        v4f a1 = __builtin_nontemporal_load(&xv[i + stride]);
        v4f a2 = __builtin_nontemporal_load(&xv[i + 2u * stride]);
        v4f a3 = __builtin_nontemporal_load(&xv[i + 3u * stride]);

        __builtin_nontemporal_store(a0 * l, &ov[i]);
        __builtin_nontemporal_store(a1 * l, &ov[i + stride]);
        __builtin_nontemporal_store(a2 * l, &ov[i + 2u * stride]);
        __builtin_nontemporal_store(a3 * l, &ov[i + 3u * stride]);
    }
    // Tail (not taken with the exact BLOCKS*TPB geometry; kept for safety).
    for (; i < n_v4; i += stride) {
        v4f a = __builtin_nontemporal_load(&xv[i]);
        __builtin_nontemporal_store(a * l, &ov[i]);
    }
}

// ---------------------------------------------------------------------------
// Launch
// ---------------------------------------------------------------------------
extern "C" void kernel_launch(void* const* d_in, const int* in_sizes, int n_in,
                              void* d_out, int out_size, void* d_ws, size_t ws_size,
                              hipStream_t stream) {
    const float* x = (const float*)d_in[0];   // [N, 2048] f32
    const float* W = (const float*)d_in[1];   // [2048, 2048] f32
    float* out = (float*)d_out;               // [N, 2048] f32
    float* lam = (float*)d_ws;                // 2048 f32 scratch (8 KB)

    // 1) gather -|diag(W)| into workspace
    lam_gather_kernel<<<(DIM + 255) / 256, 256, 0, stream>>>(W, lam);

    // 2) stream-scale: 65536*2048/4 = 33,554,432 float4 elements.
    //    BLOCKS*TPB = 524288 threads -> 64 float4 per thread, and the grid
    //    stride is a multiple of the 512-v4 row width (column invariance).
    const unsigned n_v4 = (unsigned)((long long)in_sizes[0] / 4);
    diag_scale_kernel<<<BLOCKS, TPB, 0, stream>>>(x, lam, out, n_v4);
}